// ML3Layer_18073222382240
// MI455X (gfx1250) — compile-verified
//
#include <hip/hip_runtime.h>
#include <math.h>

typedef float v2f __attribute__((ext_vector_type(2)));
typedef float v8f __attribute__((ext_vector_type(8)));

#define N_NODES 100000
#define N_EDGES 1600000
#define D_IN    128
#define D_OUT1  128
#define D_OUT2  64
#define D_TOT   192
#define E_IN    8
#define K_SPEC  4
#define M_TILES (N_NODES / 16)   // 6250, exact

// ---------------------------------------------------------------------------
// Zero the conv region of out: out[n, 0:128] = 0
// ---------------------------------------------------------------------------
__global__ void zero_conv_kernel(float* __restrict__ out) {
    int t = blockIdx.x * blockDim.x + threadIdx.x;
    if (t >= N_NODES * D_OUT1) return;
    int n = t >> 7;          // /128
    int c = t & 127;
    out[(size_t)n * D_TOT + c] = 0.0f;
}

// ---------------------------------------------------------------------------
// Edge MLP: one wave per edge.
// lane c<16  -> relu(attr . w1[:,c])            (linear part, concat idx c)
// lane c>=16 -> tanh(attr.w2[:,c-16])*tanh(attr.w3[:,c-16]) (gated, idx c)
// then ea_k = relu( sum_lane hid * w4[lane][k] ) via shfl_xor butterfly.
// ---------------------------------------------------------------------------
__global__ void edge_mlp_kernel(const float* __restrict__ edge_attr,
                                const float* __restrict__ w1,
                                const float* __restrict__ w2,
                                const float* __restrict__ w3,
                                const float* __restrict__ w4,
                                float* __restrict__ ea_out) {
    int wave = (blockIdx.x * blockDim.x + threadIdx.x) >> 5;
    int lane = threadIdx.x & 31;
    if (wave >= N_EDGES) return;

    const float* attr = edge_attr + (size_t)wave * E_IN;
    float a[E_IN];
#pragma unroll
    for (int r = 0; r < E_IN; ++r) a[r] = attr[r];

    float hid;
    if (lane < 16) {
        float s = 0.0f;
#pragma unroll
        for (int r = 0; r < E_IN; ++r) s += a[r] * w1[r * 16 + lane];
        hid = fmaxf(s, 0.0f);
    } else {
        int c = lane - 16;
        float s2 = 0.0f, s3 = 0.0f;
#pragma unroll
        for (int r = 0; r < E_IN; ++r) {
            s2 += a[r] * w2[r * 16 + c];
            s3 += a[r] * w3[r * 16 + c];
        }
        hid = tanhf(s2) * tanhf(s3);
    }

    float s[K_SPEC];
#pragma unroll
    for (int k = 0; k < K_SPEC; ++k) s[k] = hid * w4[lane * K_SPEC + k];

#pragma unroll
    for (int m = 16; m >= 1; m >>= 1) {
#pragma unroll
        for (int k = 0; k < K_SPEC; ++k) s[k] += __shfl_xor(s[k], m, 32);
    }

    if (lane < K_SPEC)
        ea_out[(size_t)wave * K_SPEC + lane] = fmaxf(s[lane], 0.0f);
}

// ---------------------------------------------------------------------------
// Conv GEMM: y[k] = x @ conv_w[k], f32 WMMA 16x16x4.
// One wave per (m-tile, k): 16 nodes x 128 output cols, K-loop over 128 feats.
// A layout (16x4 f32): lanes 0-15 -> rows M, K={0,1}; lanes 16-31 -> K={2,3}.
// B layout (4x16 f32): lanes 0-15 -> cols N, K={0,1}; lanes 16-31 -> K={2,3}.
// C/D (16x16 f32, 8 VGPRs): vgpr v, lanes 0-15 -> M=v; lanes 16-31 -> M=v+8.
// ---------------------------------------------------------------------------
__global__ void conv_gemm_kernel(const float* __restrict__ x,
                                 const float* __restrict__ conv_w,
                                 float* __restrict__ y) {
    int gw   = (blockIdx.x * blockDim.x + threadIdx.x) >> 5;
    int lane = threadIdx.x & 31;
    int k    = gw & 3;
    int mt   = gw >> 2;
    if (mt >= M_TILES) return;

    int half = lane >> 4;      // 0: K pair {0,1}, 1: K pair {2,3}
    int l16  = lane & 15;
    int arow = mt * 16 + l16;

    const float* W = conv_w + (size_t)k * D_IN * D_OUT1;

    v8f acc[8];
#pragma unroll
    for (int t = 0; t < 8; ++t) acc[t] = (v8f){};

    for (int kk = 0; kk < D_IN / 4; ++kk) {
        int f0 = kk * 4 + half * 2;
        v2f A;
        A.x = x[(size_t)arow * D_IN + f0 + 0];
        A.y = x[(size_t)arow * D_IN + f0 + 1];
#pragma unroll
        for (int t = 0; t < 8; ++t) {
            int col = t * 16 + l16;
            v2f B;
            B.x = W[(size_t)(f0 + 0) * D_OUT1 + col];
            B.y = W[(size_t)(f0 + 1) * D_OUT1 + col];
            acc[t] = __builtin_amdgcn_wmma_f32_16x16x4_f32(
                false, A, false, B, (short)0, acc[t], false, false);
        }
    }

    float* yk = y + (size_t)k * N_NODES * D_OUT1;
#pragma unroll
    for (int t = 0; t < 8; ++t) {
#pragma unroll
        for (int v = 0; v < 8; ++v) {
            int node = mt * 16 + v + half * 8;
            yk[(size_t)node * D_OUT1 + t * 16 + l16] = acc[t][v];
        }
    }
}

// ---------------------------------------------------------------------------
// Edge scatter: out[dst, 0:128] += sum_k ea[e,k] * y_k[src].
// One wave per edge; each lane handles 4 features (float4, coalesced 512B row
// reads per k); single f32 atomic per feature.
// ---------------------------------------------------------------------------
__global__ void edge_scatter_kernel(const int*   __restrict__ edge_index,
                                    const float* __restrict__ ea,
                                    const float* __restrict__ y,
                                    float*       __restrict__ out) {
    int wave = (blockIdx.x * blockDim.x + threadIdx.x) >> 5;
    int lane = threadIdx.x & 31;
    if (wave >= N_EDGES) return;

    int src = edge_index[wave];
    int dst = edge_index[N_EDGES + wave];

    float e0 = ea[(size_t)wave * 4 + 0];
    float e1 = ea[(size_t)wave * 4 + 1];
    float e2 = ea[(size_t)wave * 4 + 2];
    float e3 = ea[(size_t)wave * 4 + 3];

    int f = lane * 4;
    const size_t plane = (size_t)N_NODES * D_OUT1;
    const size_t roff  = (size_t)src * D_OUT1 + f;

    float4 v0 = *(const float4*)(y + 0 * plane + roff);
    float4 v1 = *(const float4*)(y + 1 * plane + roff);
    float4 v2 = *(const float4*)(y + 2 * plane + roff);
    float4 v3 = *(const float4*)(y + 3 * plane + roff);

    float ax = e0 * v0.x + e1 * v1.x + e2 * v2.x + e3 * v3.x;
    float ay = e0 * v0.y + e1 * v1.y + e2 * v2.y + e3 * v3.y;
    float az = e0 * v0.z + e1 * v1.z + e2 * v2.z + e3 * v3.z;
    float aw = e0 * v0.w + e1 * v1.w + e2 * v2.w + e3 * v3.w;

    float* o = out + (size_t)dst * D_TOT + f;
    __hip_atomic_fetch_add(o + 0, ax, __ATOMIC_RELAXED, __HIP_MEMORY_SCOPE_AGENT);
    __hip_atomic_fetch_add(o + 1, ay, __ATOMIC_RELAXED, __HIP_MEMORY_SCOPE_AGENT);
    __hip_atomic_fetch_add(o + 2, az, __ATOMIC_RELAXED, __HIP_MEMORY_SCOPE_AGENT);
    __hip_atomic_fetch_add(o + 3, aw, __ATOMIC_RELAXED, __HIP_MEMORY_SCOPE_AGENT);
}

// ---------------------------------------------------------------------------
// Bias + ReLU in place over conv region.
// ---------------------------------------------------------------------------
__global__ void bias_relu_kernel(float* __restrict__ out,
                                 const float* __restrict__ conv_b) {
    int t = blockIdx.x * blockDim.x + threadIdx.x;
    if (t >= N_NODES * D_OUT1) return;
    int n = t >> 7;
    int c = t & 127;
    size_t i = (size_t)n * D_TOT + c;
    out[i] = fmaxf(out[i] + conv_b[c], 0.0f);
}

// ---------------------------------------------------------------------------
// Skip branch: out[n,128:192] = tanh(x@w1+b1) * tanh(x@w2+b2), f32 WMMA.
// One wave per m-tile: 16 nodes x 64 cols, two weight matrices.
// ---------------------------------------------------------------------------
__global__ void skip_kernel(const float* __restrict__ x,
                            const float* __restrict__ w1,
                            const float* __restrict__ b1,
                            const float* __restrict__ w2,
                            const float* __restrict__ b2,
                            float* __restrict__ out) {
    int mt   = (blockIdx.x * blockDim.x + threadIdx.x) >> 5;
    int lane = threadIdx.x & 31;
    if (mt >= M_TILES) return;

    int half = lane >> 4;
    int l16  = lane & 15;
    int arow = mt * 16 + l16;

    v8f acc1[4], acc2[4];
#pragma unroll
    for (int t = 0; t < 4; ++t) { acc1[t] = (v8f){}; acc2[t] = (v8f){}; }

    for (int kk = 0; kk < D_IN / 4; ++kk) {
        int f0 = kk * 4 + half * 2;
        v2f A;
        A.x = x[(size_t)arow * D_IN + f0 + 0];
        A.y = x[(size_t)arow * D_IN + f0 + 1];
#pragma unroll
        for (int t = 0; t < 4; ++t) {
            int col = t * 16 + l16;
            v2f B1, B2;
            B1.x = w1[(size_t)(f0 + 0) * D_OUT2 + col];
            B1.y = w1[(size_t)(f0 + 1) * D_OUT2 + col];
            acc1[t] = __builtin_amdgcn_wmma_f32_16x16x4_f32(
                false, A, false, B1, (short)0, acc1[t], false, false);
            B2.x = w2[(size_t)(f0 + 0) * D_OUT2 + col];
            B2.y = w2[(size_t)(f0 + 1) * D_OUT2 + col];
            acc2[t] = __builtin_amdgcn_wmma_f32_16x16x4_f32(
                false, A, false, B2, (short)0, acc2[t], false, false);
        }
    }

#pragma unroll
    for (int t = 0; t < 4; ++t) {
        int col = t * 16 + l16;
        float bb1 = b1[col];
        float bb2 = b2[col];
#pragma unroll
        for (int v = 0; v < 8; ++v) {
            int node = mt * 16 + v + half * 8;
            float s = tanhf(acc1[t][v] + bb1) * tanhf(acc2[t][v] + bb2);
            out[(size_t)node * D_TOT + D_OUT1 + col] = s;
        }
    }
}

// ---------------------------------------------------------------------------
// Launch
// ---------------------------------------------------------------------------
extern "C" void kernel_launch(void* const* d_in, const int* in_sizes, int n_in,
                              void* d_out, int out_size, void* d_ws, size_t ws_size,
                              hipStream_t stream) {
    const float* x         = (const float*)d_in[0];
    const float* edge_attr = (const float*)d_in[1];
    const int*   edge_idx  = (const int*)  d_in[2];
    const float* e_w1      = (const float*)d_in[3];
    const float* e_w2      = (const float*)d_in[4];
    const float* e_w3      = (const float*)d_in[5];
    const float* e_w4      = (const float*)d_in[6];
    const float* conv_w    = (const float*)d_in[7];
    const float* conv_b    = (const float*)d_in[8];
    const float* skip_w1   = (const float*)d_in[9];
    const float* skip_b1   = (const float*)d_in[10];
    const float* skip_w2   = (const float*)d_in[11];
    const float* skip_b2   = (const float*)d_in[12];
    float* out = (float*)d_out;

    // workspace: ea [E,4] then y [4,N,128]
    float* ea = (float*)d_ws;
    float* y  = ea + (size_t)N_EDGES * K_SPEC;

    // 1) zero conv region of out
    {
        int total = N_NODES * D_OUT1;
        zero_conv_kernel<<<(total + 255) / 256, 256, 0, stream>>>(out);
    }
    // 2) edge MLP -> ea
    {
        long long waves = N_EDGES;
        long long blocks = (waves * 32 + 255) / 256;
        edge_mlp_kernel<<<(int)blocks, 256, 0, stream>>>(edge_attr, e_w1, e_w2,
                                                         e_w3, e_w4, ea);
    }
    // 3) y_k = x @ conv_w[k]
    {
        long long waves = (long long)M_TILES * K_SPEC;
        long long blocks = (waves * 32 + 255) / 256;
        conv_gemm_kernel<<<(int)blocks, 256, 0, stream>>>(x, conv_w, y);
    }
    // 4) edge scatter (atomics into out conv region)
    {
        long long waves = N_EDGES;
        long long blocks = (waves * 32 + 255) / 256;
        edge_scatter_kernel<<<(int)blocks, 256, 0, stream>>>(edge_idx, ea, y, out);
    }
    // 5) bias + relu in place
    {
        int total = N_NODES * D_OUT1;
        bias_relu_kernel<<<(total + 255) / 256, 256, 0, stream>>>(out, conv_b);
    }
    // 6) skip branch
    {
        long long waves = M_TILES;
        long long blocks = (waves * 32 + 255) / 256;
        skip_kernel<<<(int)blocks, 256, 0, stream>>>(x, skip_w1, skip_b1,
                                                     skip_w2, skip_b2, out);
    }
}